// DirCFConv_87445534146860
// MI455X (gfx1250) — compile-verified
//
#include <hip/hip_runtime.h>
#include <math.h>

#define LN_EPS 1e-5f

typedef __attribute__((ext_vector_type(2))) float v2f;
typedef __attribute__((ext_vector_type(4))) float v4f;
typedef __attribute__((ext_vector_type(8))) float v8f;

// ---------------------------------------------------------------------------
// Kernel 1: m = silu(layernorm(s @ W^T + b))
//   tokens T = B*N = 2048, H = 128.
//   grid = 128 blocks of 256 threads (8 wave32s).
//   Wave w computes the 16x16 output tile (rows = 16 tokens of this block,
//   cols = [16w, 16w+16)) with V_WMMA_F32_16X16X4_F32, K-loop over 128.
//   Block then owns the full 16x128 row tile -> LayerNorm over last dim in
//   LDS, SiLU, store to workspace.
// ---------------------------------------------------------------------------
__global__ __launch_bounds__(256) void lin_ln_silu_kernel(
    const float* __restrict__ s,     // (2048, 128)
    const float* __restrict__ W,     // (128, 128) row-major, h = s @ W^T
    const float* __restrict__ bias,  // (128,)
    float* __restrict__ m)           // (2048, 128) out
{
  __shared__ float tile[16][132];    // padded: conflict-free scatter + row scans
  __shared__ float s_mu[16], s_rs[16];

  const int t0   = blockIdx.x * 16;        // first token of this block
  const int lane = threadIdx.x & 31;
  const int wave = threadIdx.x >> 5;       // 0..7
  const int o0   = wave * 16;              // first output col of this wave
  const int mn   = lane & 15;              // A row / B col for this lane
  const int hi   = lane >> 4;              // lane half
  const int koff = hi * 2;                 // K offset within 4-wide slab

  // A[m][k] = s[t0+m][k];  B[k][n] = W[o0+n][k]  (since h = s @ W^T)
  const float* __restrict__ arow = s + (size_t)(t0 + mn) * 128;
  const float* __restrict__ brow = W + (size_t)(o0 + mn) * 128;

  v8f c = {};
  #pragma unroll
  for (int kk = 0; kk < 128; kk += 4) {
    v2f a, b;
    // 32-bit A 16x4 layout: lanes 0-15 hold K=kk..kk+1, lanes 16-31 K=kk+2..kk+3
    a.x = arow[kk + koff + 0];
    a.y = arow[kk + koff + 1];
    // B 4x16 mirrors the K assignment; N striped across lanes
    b.x = brow[kk + koff + 0];
    b.y = brow[kk + koff + 1];
    // 8 args: (neg_a, A, neg_b, B, c_mod, C, reuse_a, reuse_b)
    c = __builtin_amdgcn_wmma_f32_16x16x4_f32(
        false, a, false, b, (short)0, c, false, false);
  }

  // C/D layout: VGPR r, lanes 0-15 -> (row r, col lane), lanes 16-31 -> (row r+8)
  const float bv = bias[o0 + mn];
  #pragma unroll
  for (int r = 0; r < 8; ++r) {
    tile[r + hi * 8][o0 + mn] = c[r] + bv;
  }
  __syncthreads();

  // LayerNorm stats: one thread per token row (16 rows, 128 cols each)
  if (threadIdx.x < 16) {
    float sum = 0.f, sq = 0.f;
    #pragma unroll
    for (int o = 0; o < 128; ++o) {
      float v = tile[threadIdx.x][o];
      sum += v; sq += v * v;
    }
    float mu  = sum * (1.f / 128.f);
    float var = sq * (1.f / 128.f) - mu * mu;
    s_mu[threadIdx.x] = mu;
    s_rs[threadIdx.x] = rsqrtf(var + LN_EPS);
  }
  __syncthreads();

  // normalize + SiLU + store: 2048 elems / 256 threads = 8 each
  #pragma unroll
  for (int e = 0; e < 8; ++e) {
    int idx = threadIdx.x * 8 + e;       // 0..2047
    int row = idx >> 7, col = idx & 127;
    float x = (tile[row][col] - s_mu[row]) * s_rs[row];
    float act = x / (1.f + __expf(-x));  // silu
    m[(size_t)(t0 + row) * 128 + col] = act;
  }
}

// ---------------------------------------------------------------------------
// Kernel 2: out[b,i,h] = sum_j ef[b,i,j,h] * m[b,j,h]
//   Pure 512 MB stream -> HBM-bandwidth bound (~23 us at 23.3 TB/s).
//   One block per (b,i): 256 threads = 32 float4 columns x 8 j-stripes.
//   ef loads are non-temporal (read once; keep L2 for the m matrix, which is
//   re-read 512x per batch and fits easily in 192 MB L2).
// ---------------------------------------------------------------------------
__global__ __launch_bounds__(256) void efconv_kernel(
    const float* __restrict__ ef,    // (B, 512, 512, 128)
    const float* __restrict__ m,     // (B, 512, 128)
    float* __restrict__ out)         // (B, 512, 128)
{
  const int bi = blockIdx.x;          // 0..2047  (= b*512 + i)
  const int b  = bi >> 9;
  const int c4 = threadIdx.x & 31;    // float4 column within H (H/4 = 32)
  const int js = threadIdx.x >> 5;    // j-stripe 0..7

  const v4f* __restrict__ efp =
      (const v4f*)(ef + (size_t)bi * 512 * 128) + c4;          // row (b,i,j=0)
  const v4f* __restrict__ mp =
      (const v4f*)(m + (size_t)b * 512 * 128) + c4;

  v4f acc = {};
  #pragma unroll 4
  for (int j = js; j < 512; j += 8) {
    v4f e  = __builtin_nontemporal_load(efp + (size_t)j * 32);  // 32 v4f per row
    v4f mv = mp[(size_t)j * 32];
    acc += e * mv;
  }

  __shared__ v4f red[8][32];
  red[js][c4] = acc;
  __syncthreads();

  if (js == 0) {
    v4f r = red[0][c4];
    #pragma unroll
    for (int k = 1; k < 8; ++k) r += red[k][c4];
    ((v4f*)(out + (size_t)bi * 128))[c4] = r;
  }
}

// ---------------------------------------------------------------------------
// Launch: inputs are (s, ef_mask, W, b) fp32; output (B,N,H) fp32.
// d_ws holds m (2048*128 fp32 = 1 MB).
// ---------------------------------------------------------------------------
extern "C" void kernel_launch(void* const* d_in, const int* in_sizes, int n_in,
                              void* d_out, int out_size, void* d_ws, size_t ws_size,
                              hipStream_t stream) {
  const float* s    = (const float*)d_in[0];
  const float* ef   = (const float*)d_in[1];
  const float* W    = (const float*)d_in[2];
  const float* bias = (const float*)d_in[3];
  float* out = (float*)d_out;
  float* mws = (float*)d_ws;   // 2048 x 128 fp32

  lin_ln_silu_kernel<<<128, 256, 0, stream>>>(s, W, bias, mws);
  efconv_kernel<<<2048, 256, 0, stream>>>(ef, mws, out);
}